// VulnerabilityGNN_19061064860384
// MI455X (gfx1250) — compile-verified
//
#include <hip/hip_runtime.h>

typedef __attribute__((ext_vector_type(2))) float v2f;
typedef __attribute__((ext_vector_type(8))) float v8f;

// ---------------------------------------------------------------------------
// Zero-init of workspace accumulators (harness poisons ws with 0xAA once)
// ---------------------------------------------------------------------------
__global__ void zero_f32(float* __restrict__ p, long long n) {
  long long i = (long long)blockIdx.x * blockDim.x + threadIdx.x;
  long long stride = (long long)gridDim.x * blockDim.x;
  for (; i < n; i += stride) p[i] = 0.0f;
}

// ---------------------------------------------------------------------------
// Degree via f32 atomics (exact: counts << 2^24). Self-loop folded in later.
// ---------------------------------------------------------------------------
__global__ void degree_kernel(const long long* __restrict__ ei, int E,
                              float* __restrict__ deg) {
  int e = blockIdx.x * blockDim.x + threadIdx.x;
  if (e >= E) return;
  int d = (int)ei[(size_t)E + e];
  atomicAdd(&deg[d], 1.0f);
}

// dinv = rsqrt(deg + 1): self-loop makes deg >= 1 always, so no `where` needed
__global__ void dinv_kernel(const float* __restrict__ deg, float* __restrict__ dinv,
                            int N) {
  int i = blockIdx.x * blockDim.x + threadIdx.x;
  if (i >= N) return;
  dinv[i] = rsqrtf(deg[i] + 1.0f);
}

// ---------------------------------------------------------------------------
// GEMM 1:  hx[N,16] = x[N,5] @ W1[5,16]   via V_WMMA_F32_16X16X4_F32
// One wave handles a 16-node tile. K=5 -> two K=4 steps (second zero-padded).
// A layout (32-bit, 16x4): lane<16 -> K={0,1}, lane>=16 -> K={2,3}; M = lane&15
// C layout: VGPR r holds M=r (lanes 0-15) and M=r+8 (lanes 16-31), N = lane&15
// ---------------------------------------------------------------------------
__global__ void gcn_mm1_wmma(const float* __restrict__ x, const float* __restrict__ W1,
                             float* __restrict__ hx, int tiles, int N) {
  const int wave = (int)((blockIdx.x * blockDim.x + threadIdx.x) >> 5);
  const int lane = threadIdx.x & 31;
  if (wave >= tiles) return;                 // wave-uniform exit, EXEC stays all-1
  const int  lh = lane & 15;
  const bool hi = lane >= 16;
  const int  node0 = wave * 16;
  const int  row = min(node0 + lh, N - 1);   // clamp (no tail for N=100000)

  const float* xr = x + (size_t)row * 5;
  v2f a0, a1, b0, b1;
  a0.x = xr[hi ? 2 : 0];
  a0.y = xr[hi ? 3 : 1];
  a1.x = hi ? 0.0f : xr[4];                  // K=4 pad step
  a1.y = 0.0f;
  b0.x = W1[(hi ? 2 : 0) * 16 + lh];
  b0.y = W1[(hi ? 3 : 1) * 16 + lh];
  b1.x = hi ? 0.0f : W1[4 * 16 + lh];
  b1.y = 0.0f;

  v8f c = {};
  c = __builtin_amdgcn_wmma_f32_16x16x4_f32(false, a0, false, b0, (short)0, c, false, false);
  c = __builtin_amdgcn_wmma_f32_16x16x4_f32(false, a1, false, b1, (short)0, c, false, false);

  // Full-tile fast path: wave-uniform test -> straight-line stores, no exec churn
  float* o = hx + ((size_t)node0 + (hi ? 8 : 0)) * 16 + lh;
  if (node0 + 16 <= N) {
#pragma unroll
    for (int r = 0; r < 8; ++r) o[(size_t)r * 16] = c[r];
  } else {
    const int mofs = hi ? 8 : 0;
#pragma unroll
    for (int r = 0; r < 8; ++r)
      if (node0 + r + mofs < N) o[(size_t)r * 16] = c[r];
  }
}

// ---------------------------------------------------------------------------
// GEMM 2:  h2x[N,32] = h1[N,16] @ W2[16,32]  (4 K-steps x 2 N-tiles of WMMA)
// ---------------------------------------------------------------------------
__global__ void gcn_mm2_wmma(const float* __restrict__ h1, const float* __restrict__ W2,
                             float* __restrict__ h2x, int tiles, int N) {
  const int wave = (int)((blockIdx.x * blockDim.x + threadIdx.x) >> 5);
  const int lane = threadIdx.x & 31;
  if (wave >= tiles) return;
  const int  lh = lane & 15;
  const bool hi = lane >= 16;
  const int  node0 = wave * 16;
  const int  row = min(node0 + lh, N - 1);

  const float* ar = h1 + (size_t)row * 16;
  v8f c0 = {}, c1 = {};
#pragma unroll
  for (int k = 0; k < 16; k += 4) {
    const int k0 = k + (hi ? 2 : 0);
    const int k1 = k + (hi ? 3 : 1);
    v2f a, b0, b1;
    a.x  = ar[k0];
    a.y  = ar[k1];
    b0.x = W2[k0 * 32 + lh];
    b0.y = W2[k1 * 32 + lh];
    b1.x = W2[k0 * 32 + 16 + lh];
    b1.y = W2[k1 * 32 + 16 + lh];
    c0 = __builtin_amdgcn_wmma_f32_16x16x4_f32(false, a, false, b0, (short)0, c0, false, false);
    c1 = __builtin_amdgcn_wmma_f32_16x16x4_f32(false, a, false, b1, (short)0, c1, false, false);
  }

  float* o = h2x + ((size_t)node0 + (hi ? 8 : 0)) * 32 + lh;
  if (node0 + 16 <= N) {
#pragma unroll
    for (int r = 0; r < 8; ++r) {
      o[(size_t)r * 32]      = c0[r];
      o[(size_t)r * 32 + 16] = c1[r];
    }
  } else {
    const int mofs = hi ? 8 : 0;
#pragma unroll
    for (int r = 0; r < 8; ++r) {
      if (node0 + r + mofs < N) {
        o[(size_t)r * 32]      = c0[r];
        o[(size_t)r * 32 + 16] = c1[r];
      }
    }
  }
}

// ---------------------------------------------------------------------------
// Edge aggregation: agg[dst,f] += h[src,f] * dinv[src]*dinv[dst]
// Virtual edges e in [E, E+N) are the self-loops (src=dst=e-E).
// Thread = (edge, feature); feature in low bits so atomics to one dst row
// hit contiguous addresses. Tables are L2-resident; atomics resolve in L2.
// ---------------------------------------------------------------------------
template <int LOGF>
__global__ void aggregate_kernel(const long long* __restrict__ ei, int E, int N,
                                 const float* __restrict__ dinv,
                                 const float* __restrict__ h,
                                 float* __restrict__ agg) {
  const int F = 1 << LOGF;
  long long gid = (long long)blockIdx.x * blockDim.x + threadIdx.x;
  const long long total = ((long long)E + (long long)N) << LOGF;
  if (gid >= total) return;
  const int f = (int)(gid & (F - 1));
  const long long e = gid >> LOGF;
  int s, d;
  if (e < E) {
    s = (int)ei[e];
    d = (int)ei[(size_t)E + e];
  } else {
    s = d = (int)(e - E);
  }
  const float nrm = dinv[s] * dinv[d];
  atomicAdd(&agg[((size_t)d << LOGF) + f], h[((size_t)s << LOGF) + f] * nrm);
}

// h = relu(agg + bias), in place
template <int LOGF>
__global__ void bias_relu_kernel(float* __restrict__ a, const float* __restrict__ b,
                                 long long n) {
  const int F = 1 << LOGF;
  long long i = (long long)blockIdx.x * blockDim.x + threadIdx.x;
  if (i >= n) return;
  a[i] = fmaxf(a[i] + b[(int)(i & (F - 1))], 0.0f);
}

// Fused relu(agg2 + b2) + global_mean_pool numerators/denominators
__global__ void pool_kernel(const float* __restrict__ agg2, const float* __restrict__ b2,
                            const long long* __restrict__ batch,
                            float* __restrict__ pool, float* __restrict__ cnt,
                            int N) {
  long long gid = (long long)blockIdx.x * blockDim.x + threadIdx.x;
  if (gid >= (long long)N * 32) return;
  const int f = (int)(gid & 31);
  const int node = (int)(gid >> 5);
  const float v = fmaxf(agg2[gid] + b2[f], 0.0f);
  const int g = (int)batch[node];
  atomicAdd(&pool[(size_t)g * 32 + f], v);
  if (f == 0) atomicAdd(&cnt[g], 1.0f);
}

// out[g] = (pool[g,:] / max(cnt,1)) . fcW + fcb
__global__ void head_kernel(const float* __restrict__ pool, const float* __restrict__ cnt,
                            const float* __restrict__ fcW, const float* __restrict__ fcb,
                            float* __restrict__ out, int G) {
  int g = blockIdx.x * blockDim.x + threadIdx.x;
  if (g >= G) return;
  const float inv = 1.0f / fmaxf(cnt[g], 1.0f);
  float s = 0.0f;
#pragma unroll
  for (int f = 0; f < 32; ++f) s += pool[(size_t)g * 32 + f] * fcW[f];
  out[g] = s * inv + fcb[0];
}

// ---------------------------------------------------------------------------
extern "C" void kernel_launch(void* const* d_in, const int* in_sizes, int n_in,
                              void* d_out, int out_size, void* d_ws, size_t ws_size,
                              hipStream_t stream) {
  const float*     x     = (const float*)d_in[0];
  const long long* ei    = (const long long*)d_in[1];   // int64 per reference
  const long long* batch = (const long long*)d_in[2];   // int64 per reference
  const float*     W1    = (const float*)d_in[3];
  const float*     b1    = (const float*)d_in[4];
  const float*     W2    = (const float*)d_in[5];
  const float*     b2    = (const float*)d_in[6];
  const float*     fcW   = (const float*)d_in[7];
  const float*     fcb   = (const float*)d_in[8];
  float*           out   = (float*)d_out;

  const int N = in_sizes[0] / 5;   // 100000
  const int E = in_sizes[1] / 2;   // 3200000
  const int G = out_size;          // 1024

  // Workspace layout (floats). Regions needing zeros come first.
  float* ws   = (float*)d_ws;
  float* deg  = ws;                              // N        (zeroed)
  float* agg1 = deg  + N;                        // N*16     (zeroed)
  float* agg2 = agg1 + (size_t)N * 16;           // N*32     (zeroed)
  float* pool = agg2 + (size_t)N * 32;           // G*32     (zeroed)
  float* cnt  = pool + (size_t)G * 32;           // G        (zeroed)
  float* dinv = cnt  + G;                        // N
  float* hx   = dinv + N;                        // N*16
  float* h2x  = hx   + (size_t)N * 16;           // N*32

  const long long zeroN = (long long)N * 49 + (long long)G * 33;
  const int TB = 256;

  // 1) zero accumulators
  zero_f32<<<2048, TB, 0, stream>>>(ws, zeroN);

  // 2) degree + dinv
  degree_kernel<<<(E + TB - 1) / TB, TB, 0, stream>>>(ei, E, deg);
  dinv_kernel<<<(N + TB - 1) / TB, TB, 0, stream>>>(deg, dinv, N);

  // 3) hx = x @ W1 (WMMA f32 16x16x4)
  const int tiles = (N + 15) / 16;
  const int mmThreads = tiles * 32;
  gcn_mm1_wmma<<<(mmThreads + TB - 1) / TB, TB, 0, stream>>>(x, W1, hx, tiles, N);

  // 4) agg1 = sum over edges + self-loops, then h1 = relu(agg1 + b1) in place
  {
    long long tot = ((long long)E + N) << 4;
    aggregate_kernel<4><<<(int)((tot + TB - 1) / TB), TB, 0, stream>>>(ei, E, N, dinv, hx, agg1);
    long long n1 = (long long)N * 16;
    bias_relu_kernel<4><<<(int)((n1 + TB - 1) / TB), TB, 0, stream>>>(agg1, b1, n1);
  }

  // 5) h2x = h1 @ W2 (WMMA f32 16x16x4, 2 N-tiles)
  gcn_mm2_wmma<<<(mmThreads + TB - 1) / TB, TB, 0, stream>>>(agg1, W2, h2x, tiles, N);

  // 6) agg2 aggregation, then fused relu+bias+mean-pool accumulation
  {
    long long tot = ((long long)E + N) << 5;
    aggregate_kernel<5><<<(int)((tot + TB - 1) / TB), TB, 0, stream>>>(ei, E, N, dinv, h2x, agg2);
    long long np = (long long)N * 32;
    pool_kernel<<<(int)((np + TB - 1) / TB), TB, 0, stream>>>(agg2, b2, batch, pool, cnt, N);
  }

  // 7) head: out[g] = mean-pooled dot fcW + fcb
  head_kernel<<<(G + TB - 1) / TB, TB, 0, stream>>>(pool, cnt, fcW, fcb, out, G);
}